// ES_87522843560675
// MI455X (gfx1250) — compile-verified
//
#include <hip/hip_runtime.h>
#include <type_traits>

// ---------------------------------------------------------------------------
// Exponential-smoothing (Holt-Winters) forward scan, CDNA5 / gfx1250.
// Strictly serial nonlinear recurrence -> single wave32, lane 0 computes.
// CDNA5 features used: async global->LDS DMA (global_load_async_to_lds_b128,
// s_wait_asynccnt) double-buffering the input stream into LDS.
// Seasonal state (24 slots) held in registers via unroll-by-24 (static idx).
// Divisions -> v_rcp_f32 + 1 Newton step, hoisted off the serial FMA chain.
// ---------------------------------------------------------------------------

constexpr int Bq    = 512;
constexpr int Tq    = 512;
constexpr int Mq    = 24;                 // seasonality period == horizon
constexpr int TOTAL = Bq * Tq;            // 262144 steps
constexpr int CHUNK = 3072;               // staged floats per chunk (mult of 24)
constexpr int NFULL = TOTAL / CHUNK;      // 85 full chunks
constexpr int REM   = TOTAL - NFULL * CHUNK; // 1024 (= 42*24 + 16)

using TrueC  = std::integral_constant<bool, true>;
using FalseC = std::integral_constant<bool, false>;
template <int N> using IC = std::integral_constant<int, N>;

// Issue async copies: 32 lanes x 16B = 512B per instruction.
__device__ __forceinline__ void stage_async(const float* gbase, unsigned ldsBase,
                                            int nbytes) {
    unsigned lofs = ldsBase + threadIdx.x * 16u;
    const char* g = (const char*)gbase + threadIdx.x * 16u;
    for (int k = 0; k < nbytes; k += 512) {
        asm volatile("global_load_async_to_lds_b128 %0, %1, off"
                     :: "v"(lofs + (unsigned)k), "v"(g + k)
                     : "memory");
    }
}

__device__ __forceinline__ void wait_async0() {
    asm volatile("s_wait_asynccnt 0" ::: "memory");
}

__device__ __forceinline__ float rcp_nr(float x) {
    float r = __builtin_amdgcn_rcpf(x);
    return r * fmaf(-x, r, 2.0f);   // one Newton-Raphson step
}

__global__ void __launch_bounds__(32)
es_scan_kernel(const float* __restrict__ x,
               const float* __restrict__ alpha,
               const float* __restrict__ gamma,
               const float* __restrict__ s_init,
               const float* __restrict__ l_init,
               float* __restrict__ xout,    // TOTAL floats
               float2* __restrict__ out2)   // 512*24 (l, s) pairs
{
    __shared__ float sm[2][CHUNK];
    const unsigned base0 = (unsigned)(uintptr_t)&sm[0][0]; // low 32 bits = LDS offset
    const unsigned base1 = (unsigned)(uintptr_t)&sm[1][0];

    // Stage chunk 0 while we load scalars/state.
    stage_async(x, base0, CHUNK * 4);

    const float a   = alpha[0];
    const float g   = gamma[0];
    const float oma = 1.0f - a;
    const float omg = 1.0f - g;
    float level = l_init[0];

    float s[Mq], invS[Mq];
#pragma unroll
    for (int j = 0; j < Mq; ++j) {
        s[j]    = s_init[j];
        invS[j] = rcp_nr(s[j]);
    }

    const bool leader = (threadIdx.x == 0);

    // N sequential steps starting at global step tbase (tbase % 24 == 0),
    // reading from LDS at xs[0..N-1]. REC enables the horizon snapshot.
    auto steps = [&](auto recc, auto nc, int tbase, const float* xs) {
        constexpr bool REC = decltype(recc)::value;
        constexpr int  N   = decltype(nc)::value;
#pragma unroll
        for (int j = 0; j < N; ++j) {
            const int t = tbase + j;
            float y  = xs[j];
            float u  = y * invS[j];              // y / s_t  (old s, old inv)
            level    = fmaf(oma, level, a * u);  // l_t : the serial chain
            float rl = rcp_nr(level);            // 1 / l_t (off-chain)
            float sn = fmaf(g * y, rl, omg * s[j]);
            s[j]     = sn;
            invS[j]  = rcp_nr(sn);               // consumed 24 steps later
            xout[t]  = u * rl;                   // y / (s_t * l_t)
            if (REC) {
                int r = t - (Tq - 1);
                if ((unsigned)r < (unsigned)Bq) {
                    float2* o = out2 + r * Mq;
#pragma unroll
                    for (int h = 0; h < Mq; ++h)
                        o[h] = make_float2(level, s[(j + 1 + h) % Mq]);
                }
            }
        }
    };

    for (int c = 0; c < NFULL; ++c) {
        wait_async0();  // chunk c resident in sm[c&1]
        // Prefetch next chunk into the other buffer (overlaps with compute).
        if (c + 1 < NFULL)
            stage_async(x + (c + 1) * CHUNK, (c & 1) ? base0 : base1, CHUNK * 4);
        else
            stage_async(x + NFULL * CHUNK, (c & 1) ? base0 : base1, REM * 4);

        const float* xs = sm[c & 1];
        if (leader) {
            const int tbase = c * CHUNK;
            if (c == 0) {
                for (int b = 0; b < CHUNK / Mq; ++b)
                    steps(TrueC{}, IC<Mq>{}, tbase + b * Mq, xs + b * Mq);
            } else {
                for (int b = 0; b < CHUNK / Mq; ++b)
                    steps(FalseC{}, IC<Mq>{}, tbase + b * Mq, xs + b * Mq);
            }
        }
    }

    // Remainder: 1024 steps = 42 * 24 + 16, staged in sm[NFULL & 1].
    wait_async0();
    const float* xs = sm[NFULL & 1];
    if (leader) {
        const int tbase = NFULL * CHUNK;
        for (int b = 0; b < REM / Mq; ++b)
            steps(FalseC{}, IC<Mq>{}, tbase + b * Mq, xs + b * Mq);
        steps(FalseC{}, IC<REM % Mq>{}, tbase + (REM / Mq) * Mq,
              xs + (REM / Mq) * Mq);
    }
}

extern "C" void kernel_launch(void* const* d_in, const int* in_sizes, int n_in,
                              void* d_out, int out_size, void* d_ws, size_t ws_size,
                              hipStream_t stream) {
    (void)in_sizes; (void)n_in; (void)out_size; (void)d_ws; (void)ws_size;
    const float* x     = (const float*)d_in[0];
    const float* alpha = (const float*)d_in[1];
    const float* gamma = (const float*)d_in[2];
    const float* s0    = (const float*)d_in[3];
    const float* l0    = (const float*)d_in[4];
    // d_in[5] = horizon (int, ==24) is baked into Mq.

    float*  xout = (float*)d_out;
    float2* out2 = (float2*)((float*)d_out + TOTAL);

    es_scan_kernel<<<1, 32, 0, stream>>>(x, alpha, gamma, s0, l0, xout, out2);
}